// GraphEncoder_6579889897663
// MI455X (gfx1250) — compile-verified
//
#include <hip/hip_runtime.h>
#include <hip/hip_bf16.h>

// ---------------------------------------------------------------------------
// GAT encoder (3 layers) for MI455X / gfx1250, full fp32 precision.
// GEMMs use V_WMMA_F32_16X16X4_F32 (fp32 matrix cores, no downcast), with one
// wave owning a 16x128 output strip: the A fragment is loaded once per k-step
// and reused across 8 WMMAs (8 accumulators), so the loop is WMMA-dense.
// Edge phase uses fp32 atomics (global_atomic_add_f32 / ordered-int atomicMax).
// ---------------------------------------------------------------------------

typedef float v2f __attribute__((ext_vector_type(2)));
typedef float v8f __attribute__((ext_vector_type(8)));

#define DD 128          // feature dim
#define HH 8            // heads
#define DH 16           // per-head dim
#define NEG_SLOPE 0.2f

// ---- ordered-int mapping for float atomic max --------------------------------
__device__ __forceinline__ int f2ord(float f) {
    int i = __float_as_int(f);
    return i >= 0 ? i : (i ^ 0x7FFFFFFF);
}
__device__ __forceinline__ float ord2f(int i) {
    return __int_as_float(i >= 0 ? i : (i ^ 0x7FFFFFFF));
}

// ---- GEMM: C[M,128] = A[M,128] @ B[128,128], fp32 WMMA -----------------------
// One wave per 16-row strip, computing all 8 column tiles (16x128 output).
// A-tile layout (ISA 7.12.2, 32-bit A 16x4): lanes 0-15 rows, k = lhalf*2 + reg.
// B-tile layout (mirror): lanes hold columns,  k = lhalf*2 + reg.
// C/D layout: VGPR r -> row r (lanes 0-15) / row r+8 (lanes 16-31).
__global__ void gemm128_wmma_k(const float* __restrict__ A,
                               const float* __restrict__ B,
                               float* __restrict__ C, int M) {
    int gwave = (int)((blockIdx.x * blockDim.x + threadIdx.x) >> 5);
    int lane  = threadIdx.x & 31;
    int row0  = gwave << 4;
    if (row0 >= M) return;
    int lhalf = lane >> 4;          // 0 | 1
    int lmod  = lane & 15;

    const float* arow = A + (size_t)(row0 + lmod) * DD + (lhalf << 1);
    const float* brow = B + (size_t)(lhalf << 1) * DD + lmod;   // + n*16 + k*128

    v8f acc[8] = {};
#pragma unroll 2
    for (int k = 0; k < DD; k += 4) {
        v2f a;
        a.x = arow[k + 0];
        a.y = arow[k + 1];
        const float* bk = brow + (size_t)k * DD;
#pragma unroll
        for (int n = 0; n < 8; ++n) {
            v2f b;
            b.x = bk[n * DH];
            b.y = bk[n * DH + DD];
            acc[n] = __builtin_amdgcn_wmma_f32_16x16x4_f32(
                /*neg_a=*/false, a, /*neg_b=*/false, b,
                /*c_mod=*/(short)0, acc[n], /*reuse_a=*/false, /*reuse_b=*/false);
        }
    }
    float* cbase = C + (size_t)(row0 + (lhalf << 3)) * DD + lmod;
#pragma unroll
    for (int n = 0; n < 8; ++n) {
        float* crow = cbase + n * DH;
#pragma unroll
        for (int r = 0; r < 8; ++r) crow[(size_t)r * DD] = acc[n][r];
    }
}

// ---- attention logits: al[n,h] = dot(h'[n,h,:], att[h,:]) --------------------
__global__ void att_coef_k(const float* __restrict__ Hp,
                           const float* __restrict__ a_src,
                           const float* __restrict__ a_dst,
                           float* __restrict__ al_s,
                           float* __restrict__ al_d, int Nn) {
    int i = blockIdx.x * blockDim.x + threadIdx.x;   // over N*H
    if (i >= Nn * HH) return;
    int n = i >> 3, h = i & 7;
    const float* hp = Hp + (size_t)n * DD + h * DH;
    const float* as = a_src + h * DH;
    const float* ad = a_dst + h * DH;
    float ss = 0.f, sd = 0.f;
#pragma unroll
    for (int d = 0; d < DH; ++d) {
        float v = hp[d];
        ss += v * as[d];
        sd += v * ad[d];
    }
    al_s[i] = ss;
    al_d[i] = sd;
}

// ---- init kernels ------------------------------------------------------------
__global__ void init_edge_state_k(int* __restrict__ mOrd, float* __restrict__ denom, int n8) {
    int i = blockIdx.x * blockDim.x + threadIdx.x;
    if (i >= n8) return;
    mOrd[i]  = (int)0x80000000;   // < f2ord(anything finite); every dst has a self-loop
    denom[i] = 0.f;
}
__global__ void init_out_bias_k(float* __restrict__ out, const float* __restrict__ bias, int nd) {
    int i = blockIdx.x * blockDim.x + threadIdx.x;
    if (i >= nd) return;
    out[i] = bias[i & (DD - 1)];
}
__global__ void zero_stats_k(float* __restrict__ sums, float* __restrict__ sumsq) {
    int t = threadIdx.x;
    if (t < DD) { sums[t] = 0.f; sumsq[t] = 0.f; }
}

// ---- edge pass A: segment max ------------------------------------------------
__global__ void edge_max_k(const int* __restrict__ ei,
                           const float* __restrict__ als,
                           const float* __restrict__ ald,
                           int* __restrict__ mOrd, int Ecnt, int ET) {
    int i = blockIdx.x * blockDim.x + threadIdx.x;   // over ET*H
    if (i >= ET * HH) return;
    int e = i >> 3, h = i & 7;
    int s, d;
    if (e < Ecnt) { s = ei[e]; d = ei[Ecnt + e]; } else { s = d = e - Ecnt; }
    float v = als[s * HH + h] + ald[d * HH + h];
    v = v > 0.f ? v : NEG_SLOPE * v;
    atomicMax(&mOrd[d * HH + h], f2ord(v));
}

// ---- edge pass B: segment sum of exp(e - m) ----------------------------------
__global__ void edge_sum_k(const int* __restrict__ ei,
                           const float* __restrict__ als,
                           const float* __restrict__ ald,
                           const int* __restrict__ mOrd,
                           float* __restrict__ denom, int Ecnt, int ET) {
    int i = blockIdx.x * blockDim.x + threadIdx.x;
    if (i >= ET * HH) return;
    int e = i >> 3, h = i & 7;
    int s, d;
    if (e < Ecnt) { s = ei[e]; d = ei[Ecnt + e]; } else { s = d = e - Ecnt; }
    float v = als[s * HH + h] + ald[d * HH + h];
    v = v > 0.f ? v : NEG_SLOPE * v;
    float p = expf(v - ord2f(mOrd[d * HH + h]));
    atomicAdd(&denom[d * HH + h], p);
}

// ---- edge pass C: out[dst] += alpha * h'[src] (32 lanes per edge, float4) ----
__global__ void edge_scatter_k(const int* __restrict__ ei,
                               const float* __restrict__ als,
                               const float* __restrict__ ald,
                               const int* __restrict__ mOrd,
                               const float* __restrict__ denom,
                               const float* __restrict__ Hp,
                               float* __restrict__ out, int Ecnt, int ET) {
    int i = blockIdx.x * blockDim.x + threadIdx.x;   // over ET*32
    if (i >= ET * 32) return;
    int e = i >> 5, j = i & 31;      // lane j handles features j*4 .. j*4+3
    int h = j >> 2;                  // head = feature/16
    int s, d;
    if (e < Ecnt) { s = ei[e]; d = ei[Ecnt + e]; } else { s = d = e - Ecnt; }
    float v = als[s * HH + h] + ald[d * HH + h];
    v = v > 0.f ? v : NEG_SLOPE * v;
    float p     = expf(v - ord2f(mOrd[d * HH + h]));
    float alpha = p / (denom[d * HH + h] + 1e-16f);
    const float4 hv = *(const float4*)(Hp + (size_t)s * DD + j * 4);
    float* op = out + (size_t)d * DD + j * 4;
    atomicAdd(op + 0, alpha * hv.x);
    atomicAdd(op + 1, alpha * hv.y);
    atomicAdd(op + 2, alpha * hv.z);
    atomicAdd(op + 3, alpha * hv.w);
}

// ---- batch norm: column stats (sum, sumsq) -----------------------------------
__global__ void bn_stats_k(const float* __restrict__ X,
                           float* __restrict__ sums, float* __restrict__ sumsq, int Nn) {
    __shared__ float s1[256];
    __shared__ float s2[256];
    int tid = threadIdx.x;
    int col = tid & (DD - 1);
    int sub = tid >> 7;              // two rows per block-iteration
    float s = 0.f, ss = 0.f;
    for (int r = blockIdx.x * 2 + sub; r < Nn; r += gridDim.x * 2) {
        float v = X[(size_t)r * DD + col];
        s += v; ss += v * v;
    }
    s1[tid] = s; s2[tid] = ss;
    __syncthreads();
    if (sub == 0) {
        atomicAdd(&sums[col],  s1[col] + s1[col + DD]);
        atomicAdd(&sumsq[col], s2[col] + s2[col + DD]);
    }
}

// ---- batch norm: normalize ---------------------------------------------------
__global__ void bn_apply_k(const float* __restrict__ X,
                           const float* __restrict__ sums,
                           const float* __restrict__ sumsq,
                           const float* __restrict__ gamma,
                           const float* __restrict__ beta,
                           float* __restrict__ Y, int Nn) {
    int i = blockIdx.x * blockDim.x + threadIdx.x;
    if (i >= Nn * DD) return;
    int c = i & (DD - 1);
    float inv_n = 1.f / (float)Nn;
    float mean  = sums[c] * inv_n;
    float var   = sumsq[c] * inv_n - mean * mean;   // biased, as reference
    Y[i] = (X[i] - mean) * rsqrtf(var + 1e-5f) * gamma[c] + beta[c];
}

// ---------------------------------------------------------------------------
extern "C" void kernel_launch(void* const* d_in, const int* in_sizes, int n_in,
                              void* d_out, int out_size, void* d_ws, size_t ws_size,
                              hipStream_t stream) {
    const float* x       = (const float*)d_in[0];
    const int*   ei      = (const int*)d_in[1];     // [2, E] (src row, dst row)
    const float* W0      = (const float*)d_in[2];
    const float* linW    = (const float*)d_in[3];   // [3,128,128]
    const float* att_src = (const float*)d_in[4];   // [3,8,16]
    const float* att_dst = (const float*)d_in[5];
    const float* bias    = (const float*)d_in[6];   // [3,128]
    const float* gamma   = (const float*)d_in[7];   // [2,128]
    const float* beta    = (const float*)d_in[8];

    const int Nn   = in_sizes[0] / DD;   // 50000
    const int Ecnt = in_sizes[1] / 2;    // 1600000
    const int ET   = Ecnt + Nn;          // with self loops

    // workspace layout (fp32): 3 N*D buffers + 4 N*H arrays + 2*128 stats
    const size_t ND = (size_t)Nn * DD;
    float* bufH   = (float*)d_ws;        // layer input h
    float* bufHp  = bufH  + ND;          // h' = h @ W
    float* bufOut = bufHp + ND;          // aggregated output (non-final layers)
    float* al_s   = bufOut + ND;
    float* al_d   = al_s + (size_t)Nn * HH;
    int*   mOrd   = (int*)(al_d + (size_t)Nn * HH);
    float* denom  = (float*)(mOrd + (size_t)Nn * HH);
    float* sums   = denom + (size_t)Nn * HH;
    float* sumsq  = sums + DD;

    const int TPB = 256;
    // one wave per 16-row strip; 8 waves (strips) per 256-thread block
    const int gemm_blocks = ((Nn + 15) / 16 + 7) / 8;
    const int nh_blocks   = (Nn * HH + TPB - 1) / TPB;
    const int nd_blocks   = (int)((ND + TPB - 1) / TPB);
    const int e8_blocks   = (ET * HH + TPB - 1) / TPB;
    const int e32_blocks  = (int)(((size_t)ET * 32 + TPB - 1) / TPB);

    // h0 = x @ W0
    gemm128_wmma_k<<<gemm_blocks, TPB, 0, stream>>>(x, W0, bufH, Nn);

    for (int l = 0; l < 3; ++l) {
        const float* Wl = linW + (size_t)l * DD * DD;
        float* outp = (l == 2) ? (float*)d_out : bufOut;

        // h' = h @ W_l
        gemm128_wmma_k<<<gemm_blocks, TPB, 0, stream>>>(bufH, Wl, bufHp, Nn);
        // attention logits
        att_coef_k<<<nh_blocks, TPB, 0, stream>>>(bufHp, att_src + l * DD,
                                                  att_dst + l * DD, al_s, al_d, Nn);
        // init segment state + output (= bias)
        init_edge_state_k<<<nh_blocks, TPB, 0, stream>>>(mOrd, denom, Nn * HH);
        init_out_bias_k<<<nd_blocks, TPB, 0, stream>>>(outp, bias + l * DD, (int)ND);
        // softmax-over-incoming-edges, then weighted scatter
        edge_max_k<<<e8_blocks, TPB, 0, stream>>>(ei, al_s, al_d, mOrd, Ecnt, ET);
        edge_sum_k<<<e8_blocks, TPB, 0, stream>>>(ei, al_s, al_d, mOrd, denom, Ecnt, ET);
        edge_scatter_k<<<e32_blocks, TPB, 0, stream>>>(ei, al_s, al_d, mOrd, denom,
                                                       bufHp, outp, Ecnt, ET);
        if (l < 2) {
            zero_stats_k<<<1, TPB, 0, stream>>>(sums, sumsq);
            bn_stats_k<<<256, TPB, 0, stream>>>(outp, sums, sumsq, Nn);
            bn_apply_k<<<nd_blocks, TPB, 0, stream>>>(outp, sums, sumsq,
                                                      gamma + l * DD, beta + l * DD,
                                                      bufH, Nn);
        }
    }
}